// SparseSelfAttention_61418032332869
// MI455X (gfx1250) — compile-verified
//
#include <hip/hip_runtime.h>
#include <cstddef>

// ---------------------------------------------------------------------------
// MI455X (gfx1250) fused sparse self-attention.
// Precision: fp32 end-to-end via V_WMMA_F32_16X16X4_F32 (problem is HBM-bound
// on the 270MB allattn output; cheaper WMMA dtypes would not help).
// ---------------------------------------------------------------------------

typedef __attribute__((ext_vector_type(2))) float v2f;
typedef __attribute__((ext_vector_type(8))) float v8f;

#define B_SZ    2
#define S_LEN   4096
#define DM      512
#define NHEAD   8
#define DK      64
#define RATE    4
#define KPATCH  7
#define N_PAD   4100            // S + pad_len (pad_len = 4)
#define L_LEN   1025            // N_PAD / RATE
#define ACOLS   1032            // L_LEN + KPATCH
#define M_ROWS  (B_SZ * N_PAD)  // 8200 projected rows
#define OUT0    ((size_t)B_SZ * S_LEN * DM)  // elements of `out` before allattn
#define SCALE   0.125f          // 1/sqrt(DK)
#define SC_STR  1051            // score-row stride in LDS (odd-ish vs 64 banks)
#define LOCAL0  1040            // column offset of the 7 local-patch scores

// fp32 WMMA: D(16x16) = A(16x4) * B(4x16) + C.  wave32 operand striping:
//   A: lane l holds A[M = l%16][K = 2*(l/16) + j] in vgpr j   (j = 0,1)
//   B: lane l holds B[K = 2*(l/16) + j][N = l%16] in vgpr j
//   C/D: vgpr v, lane l -> row v + 8*(l/16), col l%16
static __device__ __forceinline__ v8f wmma4(v2f a, v2f b, v8f c) {
  return __builtin_amdgcn_wmma_f32_16x16x4_f32(false, a, false, b, (short)0, c,
                                               false, false);
}

// ---------------------------------------------------------------------------
// Kernel 1: C[8200,512] = srcp[8200,512] @ W[512,512] for W in {Wq,Wk,Wv}.
// srcp rows with (row % 4100) >= 4096 are the zero pad rows.
// Workgroup = 128 threads (4 waves), 64x64 output tile; wave w owns rows
// [16w,16w+16) x 64 cols (four 16x16 WMMA accumulators).
// ---------------------------------------------------------------------------
__global__ __launch_bounds__(128) void qkv_gemm(
    const float* __restrict__ src, const float* __restrict__ Wq,
    const float* __restrict__ Wk, const float* __restrict__ Wv,
    float* __restrict__ qo, float* __restrict__ ko, float* __restrict__ vo) {
  const float* W;
  float* out;
  if (blockIdx.z == 0)      { W = Wq; out = qo; }
  else if (blockIdx.z == 1) { W = Wk; out = ko; }
  else                      { W = Wv; out = vo; }

  __shared__ float As[64][17];   // 64 rows x 16 k   (stride 17: conflict-free)
  __shared__ float Bs[16][68];   // 16 k    x 64 n

  const int tid  = threadIdx.x;
  const int wave = tid >> 5;
  const int lane = tid & 31;
  const int m0   = blockIdx.x * 64;
  const int n0   = blockIdx.y * 64;

  v8f acc0 = {}, acc1 = {}, acc2 = {}, acc3 = {};

  const int arow  = wave * 16 + (lane & 15);
  const int khalf = 2 * (lane >> 4);

  for (int kb = 0; kb < DM; kb += 16) {
    __syncthreads();
    {  // stage A: 64x16 (2 float4 per thread)
      int row = tid >> 1;
      int cb  = (tid & 1) * 8;
      int gr  = m0 + row;
      int t   = gr % N_PAD;
      if (gr < M_ROWS && t < S_LEN) {
        const float* p = src + ((size_t)(gr / N_PAD) * S_LEN + t) * DM + kb + cb;
        float4 x0 = *(const float4*)(p);
        float4 x1 = *(const float4*)(p + 4);
        As[row][cb + 0] = x0.x; As[row][cb + 1] = x0.y;
        As[row][cb + 2] = x0.z; As[row][cb + 3] = x0.w;
        As[row][cb + 4] = x1.x; As[row][cb + 5] = x1.y;
        As[row][cb + 6] = x1.z; As[row][cb + 7] = x1.w;
      } else {
#pragma unroll
        for (int j = 0; j < 8; ++j) As[row][cb + j] = 0.0f;
      }
    }
    {  // stage W tile: 16x64
      int row = tid >> 3;
      int cb  = (tid & 7) * 8;
      const float* p = W + (size_t)(kb + row) * DM + n0 + cb;
      float4 x0 = *(const float4*)(p);
      float4 x1 = *(const float4*)(p + 4);
      Bs[row][cb + 0] = x0.x; Bs[row][cb + 1] = x0.y;
      Bs[row][cb + 2] = x0.z; Bs[row][cb + 3] = x0.w;
      Bs[row][cb + 4] = x1.x; Bs[row][cb + 5] = x1.y;
      Bs[row][cb + 6] = x1.z; Bs[row][cb + 7] = x1.w;
    }
    __syncthreads();

#pragma unroll
    for (int ks = 0; ks < 16; ks += 4) {
      v2f a;
      a.x = As[arow][ks + khalf];
      a.y = As[arow][ks + khalf + 1];
      v2f b0, b1, b2, b3;
      const int nl = lane & 15;
      b0.x = Bs[ks + khalf][nl +  0]; b0.y = Bs[ks + khalf + 1][nl +  0];
      b1.x = Bs[ks + khalf][nl + 16]; b1.y = Bs[ks + khalf + 1][nl + 16];
      b2.x = Bs[ks + khalf][nl + 32]; b2.y = Bs[ks + khalf + 1][nl + 32];
      b3.x = Bs[ks + khalf][nl + 48]; b3.y = Bs[ks + khalf + 1][nl + 48];
      acc0 = wmma4(a, b0, acc0);
      acc1 = wmma4(a, b1, acc1);
      acc2 = wmma4(a, b2, acc2);
      acc3 = wmma4(a, b3, acc3);
    }
  }

  const int orow = m0 + wave * 16 + 8 * (lane >> 4);
  const int ocol = n0 + (lane & 15);
#pragma unroll
  for (int v = 0; v < 8; ++v) {
    int gr = orow + v;
    if (gr < M_ROWS) {
      float* p = out + (size_t)gr * DM + ocol;
      p[0]  = acc0[v];
      p[16] = acc1[v];
      p[32] = acc2[v];
      p[48] = acc3[v];
    }
  }
}

// ---------------------------------------------------------------------------
// Kernel 2: fused attention per (b,h,r, 16-row l-tile).
// Phases: stage Q + local K/V windows -> local-patch scores (VALU) ->
// QK^T scores (WMMA, waves split the 65 column tiles) -> row softmax
// (writes allattn) -> P@V (WMMA, waves split the 64 d-columns) + local term
// -> store out.
// ---------------------------------------------------------------------------
__global__ __launch_bounds__(128) void attn_fused(
    const float* __restrict__ qb, const float* __restrict__ kb,
    const float* __restrict__ vb, float* __restrict__ dout) {
  __shared__ float Qs[16][68];        // Q tile [16 rows][64 d]
  __shared__ float Sc[16][SC_STR];    // scores: 0..1039 atr(+pad0), 1040..1046 local
  __shared__ float Kwin[68][68];      // 67-position local K window
  __shared__ float Vwin[68][68];      // 67-position local V window
  __shared__ float St[4][16][68];     // per-wave K tile stage / shared V tile stage
  __shared__ float Red[16][8];        // softmax reduction scratch

  const int tid  = threadIdx.x;
  const int wave = tid >> 5;
  const int lane = tid & 31;
  const int lt   = blockIdx.x;        // 0..64 (16-row tiles of L=1025)
  const int bhr  = blockIdx.y;        // 0..63
  const int b    = bhr >> 5;
  const int h    = (bhr >> 2) & 7;
  const int r    = bhr & 3;
  const int l0   = lt * 16;

  const size_t headoff = (size_t)h * DK;
  const float* qbase = qb + (size_t)b * N_PAD * DM + headoff;
  const float* kbase = kb + (size_t)b * N_PAD * DM + headoff;
  const float* vbase = vb + (size_t)b * N_PAD * DM + headoff;

  // ---- Phase 0: stage Q tile and the 67-row local K/V windows ----
  for (int idx = tid; idx < 16 * 16; idx += 128) {
    int i = idx >> 4, cb = (idx & 15) * 4;
    int l = l0 + i;
    if (l < L_LEN) {
      float4 x = *(const float4*)(qbase + (size_t)(l * RATE + r) * DM + cb);
      Qs[i][cb] = x.x; Qs[i][cb + 1] = x.y; Qs[i][cb + 2] = x.z; Qs[i][cb + 3] = x.w;
    } else {
      Qs[i][cb] = 0.f; Qs[i][cb + 1] = 0.f; Qs[i][cb + 2] = 0.f; Qs[i][cb + 3] = 0.f;
    }
  }
  const int winStart = l0 * RATE + r - 3;
  for (int idx = tid; idx < 67 * 16; idx += 128) {
    int w = idx >> 4, cb = (idx & 15) * 4;
    int p = winStart + w;
    float4 xk = make_float4(0.f, 0.f, 0.f, 0.f);
    float4 xv = make_float4(0.f, 0.f, 0.f, 0.f);
    if (p >= 0 && p < N_PAD) {
      xk = *(const float4*)(kbase + (size_t)p * DM + cb);
      xv = *(const float4*)(vbase + (size_t)p * DM + cb);
    }
    Kwin[w][cb] = xk.x; Kwin[w][cb + 1] = xk.y; Kwin[w][cb + 2] = xk.z; Kwin[w][cb + 3] = xk.w;
    Vwin[w][cb] = xv.x; Vwin[w][cb + 1] = xv.y; Vwin[w][cb + 2] = xv.z; Vwin[w][cb + 3] = xv.w;
  }
  __syncthreads();

  // ---- Phase 1: 7 local patch scores per row (112 small dot products) ----
  if (tid < 16 * KPATCH) {
    int i = tid / KPATCH, j = tid % KPATCH;
    float s = 0.f;
    for (int d = 0; d < DK; ++d) s += Qs[i][d] * Kwin[i * RATE + j][d];
    Sc[i][LOCAL0 + j] = s * SCALE;
  }

  // ---- Phase 2: atr scores  S[16 x 1025] = Q K^T * scale  (WMMA) ----
  const int arow  = lane & 15;
  const int khalf = 2 * (lane >> 4);
  for (int mt = wave; mt < 65; mt += 4) {
    // wave-private stage of K tile [16 cols(m)][64 d]
    for (int idx = lane; idx < 16 * 16; idx += 32) {
      int kk = idx >> 4, cb = (idx & 15) * 4;
      int mm = mt * 16 + kk;
      float4 x = make_float4(0.f, 0.f, 0.f, 0.f);
      if (mm < L_LEN) x = *(const float4*)(kbase + (size_t)(mm * RATE + r) * DM + cb);
      St[wave][kk][cb] = x.x; St[wave][kk][cb + 1] = x.y;
      St[wave][kk][cb + 2] = x.z; St[wave][kk][cb + 3] = x.w;
    }
    v8f acc = {};
#pragma unroll
    for (int ks = 0; ks < DK; ks += 4) {
      v2f a, bo;
      a.x  = Qs[arow][ks + khalf];
      a.y  = Qs[arow][ks + khalf + 1];
      bo.x = St[wave][arow][ks + khalf];      // B[K=d][N=m] = Ktile[m][d]
      bo.y = St[wave][arow][ks + khalf + 1];
      acc = wmma4(a, bo, acc);
    }
#pragma unroll
    for (int v = 0; v < 8; ++v) {
      int i = v + 8 * (lane >> 4);
      int col = mt * 16 + (lane & 15);
      Sc[i][col] = (col < L_LEN) ? acc[v] * SCALE : 0.0f;
    }
  }
  __syncthreads();

  // ---- Phase 3: softmax over 1032 valid cols; write allattn ----
  {
    const int i = tid >> 3;   // row (8 lanes per row, all intra-wave)
    const int sub = tid & 7;
    float m = -3.0e38f;
    for (int c = sub; c < L_LEN; c += 8) m = fmaxf(m, Sc[i][c]);
    if (sub < KPATCH) m = fmaxf(m, Sc[i][LOCAL0 + sub]);
    Red[i][sub] = m;
    float rowmax = Red[i][0];
#pragma unroll
    for (int j = 1; j < 8; ++j) rowmax = fmaxf(rowmax, Red[i][j]);

    float s = 0.f;
    for (int c = sub; c < L_LEN; c += 8) s += __expf(Sc[i][c] - rowmax);
    if (sub < KPATCH) s += __expf(Sc[i][LOCAL0 + sub] - rowmax);
    Red[i][sub] = s;
    float tot = 0.f;
#pragma unroll
    for (int j = 0; j < 8; ++j) tot += Red[i][j];
    const float inv = 1.0f / tot;

    const int l = l0 + i;
    float* arow_out = dout + OUT0 +
        ((((size_t)b * NHEAD + h) * RATE + r) * L_LEN + (size_t)l) * ACOLS;
    for (int c = sub; c < 1040; c += 8) {
      float p = (c < L_LEN) ? __expf(Sc[i][c] - rowmax) * inv : 0.0f;
      Sc[i][c] = p;                       // zero-padded cols feed the PV GEMM
      if (c < L_LEN && l < L_LEN) arow_out[c] = p;
    }
    if (sub < KPATCH) {
      float p = __expf(Sc[i][LOCAL0 + sub] - rowmax) * inv;
      Sc[i][LOCAL0 + sub] = p;
      if (l < L_LEN) arow_out[L_LEN + sub] = p;
    }
  }
  __syncthreads();

  // ---- Phase 4: out[16x64] = P[16x1040] @ V[1040x64]  (WMMA) ----
  v8f oacc = {};
  const int d0 = wave * 16;
  for (int mt = 0; mt < 65; ++mt) {
    __syncthreads();  // previous tile fully consumed before restage
    for (int idx = tid; idx < 16 * 16; idx += 128) {
      int kk = idx >> 4, cb = (idx & 15) * 4;
      int mm = mt * 16 + kk;
      float4 x = make_float4(0.f, 0.f, 0.f, 0.f);
      if (mm < L_LEN) x = *(const float4*)(vbase + (size_t)(mm * RATE + r) * DM + cb);
      St[0][kk][cb] = x.x; St[0][kk][cb + 1] = x.y;
      St[0][kk][cb + 2] = x.z; St[0][kk][cb + 3] = x.w;
    }
    __syncthreads();
#pragma unroll
    for (int ks = 0; ks < 16; ks += 4) {
      v2f a, bo;
      const int kcol = mt * 16 + ks + khalf;
      a.x  = Sc[arow][kcol];
      a.y  = Sc[arow][kcol + 1];
      bo.x = St[0][ks + khalf][d0 + arow];     // B[K=m][N=d] = Vtile[m][d]
      bo.y = St[0][ks + khalf + 1][d0 + arow];
      oacc = wmma4(a, bo, oacc);
    }
  }

  // local patch contribution + store out (drop the 4 padded positions)
#pragma unroll
  for (int v = 0; v < 8; ++v) {
    int i = v + 8 * (lane >> 4);
    int n = lane & 15;
    float sum = oacc[v];
#pragma unroll
    for (int j = 0; j < KPATCH; ++j)
      sum += Sc[i][LOCAL0 + j] * Vwin[i * RATE + j][d0 + n];
    int t = (l0 + i) * RATE + r;
    if (t < S_LEN)
      dout[((size_t)b * S_LEN + t) * DM + headoff + d0 + n] = sum;
  }
}

// ---------------------------------------------------------------------------
extern "C" void kernel_launch(void* const* d_in, const int* in_sizes, int n_in,
                              void* d_out, int out_size, void* d_ws,
                              size_t ws_size, hipStream_t stream) {
  (void)in_sizes; (void)n_in; (void)out_size; (void)ws_size;
  const float* src = (const float*)d_in[0];
  const float* Wq  = (const float*)d_in[1];
  const float* Wk  = (const float*)d_in[2];
  const float* Wv  = (const float*)d_in[3];
  float* out = (float*)d_out;

  float* qbuf = (float*)d_ws;                          // [8200 x 512] each
  float* kbuf = qbuf + (size_t)M_ROWS * DM;
  float* vbuf = kbuf + (size_t)M_ROWS * DM;

  dim3 g1((M_ROWS + 63) / 64, DM / 64, 3);
  qkv_gemm<<<g1, 128, 0, stream>>>(src, Wq, Wk, Wv, qbuf, kbuf, vbuf);

  dim3 g2(65, B_SZ * NHEAD * RATE);                    // (l-tiles, b*h*r)
  attn_fused<<<g2, 128, 0, stream>>>(qbuf, kbuf, vbuf, out);
}